// Theta_Estimation_63050119905742
// MI455X (gfx1250) — compile-verified
//
#include <hip/hip_runtime.h>
#include <hip/hip_bf16.h>

#define N_Q    8192
#define DIM    384
#define M_DOCS 1024
#define KC     64
#define LDSS   68   // LDS row stride (floats): avoids bank conflicts, keeps 8B align

typedef __attribute__((ext_vector_type(2))) float v2f;
typedef __attribute__((ext_vector_type(4))) float f32x4;
typedef __attribute__((ext_vector_type(4))) int   i32x4;
typedef __attribute__((ext_vector_type(8))) float v8f;

// ---------------- zero H/den scratch ----------------
__global__ void k_zero(float* __restrict__ p, int n) {
    int i = blockIdx.x * blockDim.x + threadIdx.x;
    if (i < n) p[i] = 0.0f;
}

// ---------------- row-normalize emb_q ----------------
__global__ __launch_bounds__(128) void k_normalize(const float* __restrict__ emb,
                                                   float* __restrict__ nrm) {
    const int row = blockIdx.x;
    const int t   = threadIdx.x;                  // 128 threads, 3 elems each
    const float* x = emb + (size_t)row * DIM;
    float*       y = nrm + (size_t)row * DIM;
    float v0 = x[t], v1 = x[t + 128], v2 = x[t + 256];
    float s = v0 * v0 + v1 * v1 + v2 * v2;
    #pragma unroll
    for (int off = 16; off > 0; off >>= 1) s += __shfl_xor(s, off, 32);
    __shared__ float ws[4];
    __shared__ float rn;
    if ((t & 31) == 0) ws[t >> 5] = s;
    __syncthreads();
    if (t == 0) rn = 1.0f / sqrtf(ws[0] + ws[1] + ws[2] + ws[3]);
    __syncthreads();
    const float r = rn;
    y[t] = v0 * r; y[t + 128] = v1 * r; y[t + 256] = v2 * r;
}

// ---------------- segment scatter-add: H[m] += label*n_q, den[m] += label ----------------
__global__ __launch_bounds__(128) void k_segsum(const float* __restrict__ nrm,
                                                const float* __restrict__ label,
                                                const int*   __restrict__ doc,
                                                float* __restrict__ H,
                                                float* __restrict__ den) {
    const int q   = blockIdx.x;
    const int t   = threadIdx.x;
    const int m   = doc[q];
    const float l = label[q];
    const float* x = nrm + (size_t)q * DIM;
    float*       h = H   + (size_t)m * DIM;
    atomicAdd(&h[t],        l * x[t]);
    atomicAdd(&h[t + 128],  l * x[t + 128]);
    atomicAdd(&h[t + 256],  l * x[t + 256]);
    if (t == 0) atomicAdd(&den[m], l);
}

// ---------------- H[m] /= den[m] ----------------
__global__ __launch_bounds__(128) void k_scale(float* __restrict__ H,
                                               const float* __restrict__ den) {
    const int m = blockIdx.x;
    const int t = threadIdx.x;
    const float d  = den[m];
    const float iv = (d != 0.0f) ? (1.0f / d) : 0.0f;
    float* h = H + (size_t)m * DIM;
    h[t] *= iv; h[t + 128] *= iv; h[t + 256] *= iv;
}

// ---------------- P = nrm @ H^T  via V_WMMA_F32_16X16X4_F32 ----------------
// Block: 256 threads (8 waves), 64x64 output tile. Wave w: row-tile (w&3),
// col-tiles 2*(w>>2)+{0,1}. A/B staged through LDS with padded stride.
__global__ __launch_bounds__(256) void k_gemm(const float* __restrict__ A,   // [N_Q, DIM]
                                              const float* __restrict__ B,   // [M_DOCS, DIM]
                                              float* __restrict__ P) {       // [N_Q, M_DOCS]
    __shared__ float sA[64 * LDSS];
    __shared__ float sB[64 * LDSS];

    const int t     = threadIdx.x;
    const int w     = t >> 5;
    const int lane  = t & 31;
    const int mlane = lane & 15;     // row within 16x16 tile (A), col (B/C)
    const int half  = lane >> 4;     // selects K pair {0,1} vs {2,3}; C rows +8
    const int rt    = w & 3;         // row tile 0..3
    const int ct0   = (w >> 2) * 2;  // first of two col tiles

    const int rowA = blockIdx.y * 64;
    const int colB = blockIdx.x * 64;

    v8f acc0 = {};
    v8f acc1 = {};

    for (int kt = 0; kt < DIM / KC; ++kt) {
        #pragma unroll
        for (int i = 0; i < 4; ++i) {
            const int idx = t + i * 256;
            const int r   = idx >> 4;          // 0..63
            const int c4  = (idx & 15) * 4;    // 0..60
            const f32x4 va = *(const f32x4*)(A + (size_t)(rowA + r) * DIM + kt * KC + c4);
            *(f32x4*)(&sA[r * LDSS + c4]) = va;
            const f32x4 vb = *(const f32x4*)(B + (size_t)(colB + r) * DIM + kt * KC + c4);
            *(f32x4*)(&sB[r * LDSS + c4]) = vb;
        }
        __syncthreads();

        const float* aB  = &sA[(rt * 16        + mlane) * LDSS + half * 2];
        const float* b0B = &sB[(ct0 * 16       + mlane) * LDSS + half * 2];
        const float* b1B = &sB[((ct0 + 1) * 16 + mlane) * LDSS + half * 2];

        #pragma unroll
        for (int kk = 0; kk < KC / 4; ++kk) {
            const v2f a  = *(const v2f*)(aB  + kk * 4);
            const v2f b0 = *(const v2f*)(b0B + kk * 4);
            const v2f b1 = *(const v2f*)(b1B + kk * 4);
            acc0 = __builtin_amdgcn_wmma_f32_16x16x4_f32(false, a, false, b0,
                                                         (short)0, acc0, false, false);
            acc1 = __builtin_amdgcn_wmma_f32_16x16x4_f32(false, a, false, b1,
                                                         (short)0, acc1, false, false);
        }
        __syncthreads();
    }

    // C/D layout: lane n = lane&15; VGPR v holds row v + 8*(lane>>4)
    const int rowBase = rowA + rt * 16 + half * 8;
    const int col0 = colB + ct0 * 16 + mlane;
    const int col1 = colB + (ct0 + 1) * 16 + mlane;
    #pragma unroll
    for (int v = 0; v < 8; ++v) {
        P[(size_t)(rowBase + v) * M_DOCS + col0] = acc0[v];
        P[(size_t)(rowBase + v) * M_DOCS + col1] = acc1[v];
    }
}

// ---------------- theta[i,q] = P[i, doc[q]]  (268 MB stream, NT stores) ----------------
__global__ __launch_bounds__(256) void k_gather(const float* __restrict__ P,
                                                const int*   __restrict__ doc,
                                                float* __restrict__ out) {
    const int t     = threadIdx.x;
    const int qbase = blockIdx.x * 1024;     // 256 threads x 4 q each
    const int rbase = blockIdx.y * 16;       // 16 rows per block: reuse regs for ids
    const i32x4 d = *(const i32x4*)(doc + qbase + t * 4);
    #pragma unroll
    for (int r = 0; r < 16; ++r) {
        const int row = rbase + r;
        const float* prow = P + (size_t)row * M_DOCS;
        f32x4 v;
        v.x = prow[d.x]; v.y = prow[d.y]; v.z = prow[d.z]; v.w = prow[d.w];
        __builtin_nontemporal_store(v, (f32x4*)(out + (size_t)row * N_Q + qbase + t * 4));
    }
}

extern "C" void kernel_launch(void* const* d_in, const int* in_sizes, int n_in,
                              void* d_out, int out_size, void* d_ws, size_t ws_size,
                              hipStream_t stream) {
    const float* emb   = (const float*)d_in[0];
    const float* label = (const float*)d_in[1];
    const int*   doc   = (const int*)d_in[2];
    float* out = (float*)d_out;

    // workspace layout (all 16B aligned): norm | H | den | P  (~48 MB)
    float* nrm = (float*)d_ws;
    float* H   = nrm + (size_t)N_Q * DIM;
    float* den = H   + (size_t)M_DOCS * DIM;
    float* P   = den + M_DOCS;

    const int nz = M_DOCS * DIM + M_DOCS;   // H and den are contiguous
    k_zero<<<dim3((nz + 255) / 256), dim3(256), 0, stream>>>(H, nz);
    k_normalize<<<dim3(N_Q), dim3(128), 0, stream>>>(emb, nrm);
    k_segsum<<<dim3(N_Q), dim3(128), 0, stream>>>(nrm, label, doc, H, den);
    k_scale<<<dim3(M_DOCS), dim3(128), 0, stream>>>(H, den);
    k_gemm<<<dim3(M_DOCS / 64, N_Q / 64), dim3(256), 0, stream>>>(nrm, H, P);
    k_gather<<<dim3(N_Q / 1024, N_Q / 16), dim3(256), 0, stream>>>(P, doc, out);
}